// GCN_42760694399684
// MI455X (gfx1250) — compile-verified
//
#include <hip/hip_runtime.h>
#include <hip/hip_bf16.h>
#include <math.h>

#define N_NODES 100000
#define N_EDGES 1600000
#define N_GRAPHS 64
#define FDIM 128
#define GF 8

typedef float v2f __attribute__((ext_vector_type(2)));
typedef float v8f __attribute__((ext_vector_type(8)));

__device__ __forceinline__ void atomAddF(float* p, float v) {
    unsafeAtomicAdd(p, v);   // native GLOBAL_ATOMIC_ADD_F32
}

// ---------------------------------------------------------------------------
// Generic fills
// ---------------------------------------------------------------------------
__global__ void zerof_kernel(float* __restrict__ p, int n) {
    int i = blockIdx.x * blockDim.x + threadIdx.x;
    if (i < n) p[i] = 0.0f;
}
__global__ void zeroi_kernel(int* __restrict__ p, int n) {
    int i = blockIdx.x * blockDim.x + threadIdx.x;
    if (i < n) p[i] = 0;
}

// ---------------------------------------------------------------------------
// In-degree histogram (int) -> dinv = rsqrt(deg+1)
// ---------------------------------------------------------------------------
__global__ void deg_count_kernel(const int* __restrict__ dst, int* __restrict__ counts, int nE) {
    int e = blockIdx.x * blockDim.x + threadIdx.x;
    if (e < nE) atomicAdd(&counts[dst[e]], 1);
}

__global__ void dinv_kernel(const int* __restrict__ counts, float* __restrict__ dinv, int n) {
    int i = blockIdx.x * blockDim.x + threadIdx.x;
    if (i < n) dinv[i] = rsqrtf((float)counts[i] + 1.0f);   // +1 self-loop
}

// ---------------------------------------------------------------------------
// Single-workgroup chunked exclusive scan (1024 threads = 32 waves).
// counts[0..n) -> offsets[0..n] (offsets[n] = total), cursor = copy of offsets.
// Runs once; 100K elements -> negligible.
// ---------------------------------------------------------------------------
__global__ void __launch_bounds__(1024)
scan_kernel(const int* __restrict__ counts, int* __restrict__ offsets,
            int* __restrict__ cursor, int n) {
    __shared__ int buf[1024];
    __shared__ int carry;
    const int tid = threadIdx.x;
    if (tid == 0) carry = 0;
    __syncthreads();
    for (int base = 0; base < n; base += 1024) {
        int i = base + tid;
        int v = (i < n) ? counts[i] : 0;
        int c0 = carry;
        __syncthreads();
        buf[tid] = v;
        __syncthreads();
        for (int ofs = 1; ofs < 1024; ofs <<= 1) {
            int t = (tid >= ofs) ? buf[tid - ofs] : 0;
            __syncthreads();
            buf[tid] += t;
            __syncthreads();
        }
        int excl = c0 + buf[tid] - v;
        if (i < n) { offsets[i] = excl; cursor[i] = excl; }
        if (tid == 1023) carry = c0 + buf[1023];
        __syncthreads();
    }
    if (tid == 0) offsets[n] = carry;
}

// ---------------------------------------------------------------------------
// CSR build (once): slot via native int atomic; store source node + edge norm.
// ---------------------------------------------------------------------------
__global__ void csr_build_kernel(const int* __restrict__ src, const int* __restrict__ dst,
                                 const float* __restrict__ dinv, int* __restrict__ cursor,
                                 int* __restrict__ csr_src, float* __restrict__ csr_nrm, int nE) {
    int e = blockIdx.x * blockDim.x + threadIdx.x;
    if (e < nE) {
        int s = src[e], d = dst[e];
        int slot = atomicAdd(&cursor[d], 1);
        csr_src[slot] = s;
        csr_nrm[slot] = dinv[s] * dinv[d];
    }
}

// ---------------------------------------------------------------------------
// GEMM: out[N,128] = A[N,128] @ W[128,128], f32 WMMA 16x16x4.
// Block = 256 threads (8 waves); block owns 16 rows, wave w owns cols [16w,16w+16).
// VGPR layouts per CDNA5 ISA 7.12.2 (verified emission round 1).
// ---------------------------------------------------------------------------
#define AS 132

__global__ void __launch_bounds__(256)
gemm128_wmma_kernel(const float* __restrict__ A, const float* __restrict__ W,
                    float* __restrict__ out) {
    __shared__ float lA[16 * AS];
    const int tid  = threadIdx.x;
    const int wave = tid >> 5;
    const int lane = tid & 31;
    const int half = lane >> 4;
    const int l16  = lane & 15;
    const size_t rowbase = (size_t)blockIdx.x * 16;

    for (int t = tid; t < 16 * FDIM; t += 256) {
        int r = t >> 7, c = t & 127;
        lA[r * AS + c] = A[(rowbase + r) * FDIM + c];
    }
    __syncthreads();

    const int colbase = wave * 16;
    v8f acc = {};
    #pragma unroll 4
    for (int k = 0; k < FDIM; k += 4) {
        const int ka = k + half * 2;
        v2f a, b;
        a.x = lA[l16 * AS + ka];
        a.y = lA[l16 * AS + ka + 1];
        b.x = W[(size_t)(ka + 0) * FDIM + colbase + l16];
        b.y = W[(size_t)(ka + 1) * FDIM + colbase + l16];
        acc = __builtin_amdgcn_wmma_f32_16x16x4_f32(
            false, a, false, b, (short)0, acc, false, false);
    }

    #pragma unroll
    for (int r = 0; r < 8; ++r) {
        int m = r + half * 8;
        out[(rowbase + m) * FDIM + colbase + l16] = acc[r];
    }
}

// ---------------------------------------------------------------------------
// Fused per-layer aggregation (atomic-free):
//   h[n] = act( sum_{j in CSR[n]} xw[src_j]*nrm_j  +  xw[n]*dinv[n]^2  +  bias )
// One wave per node; lane handles 4 features (float4). csr entries are
// wave-uniform loads (single broadcast request); xw gathers are L2-resident.
// ---------------------------------------------------------------------------
__global__ void __launch_bounds__(256)
agg_fused_kernel(const float* __restrict__ xw, const int* __restrict__ offsets,
                 const int* __restrict__ csr_src, const float* __restrict__ csr_nrm,
                 const float* __restrict__ dinv, const float* __restrict__ bias,
                 float* __restrict__ h, int do_elu, int n) {
    int gid  = blockIdx.x * blockDim.x + threadIdx.x;
    int node = gid >> 5;
    int lane = gid & 31;
    if (node >= n) return;

    // self-loop seed
    float di = dinv[node];
    float s2 = di * di;
    float4 acc = ((const float4*)(xw + (size_t)node * FDIM))[lane];
    acc.x *= s2; acc.y *= s2; acc.z *= s2; acc.w *= s2;

    const int jb = offsets[node];
    const int je = offsets[node + 1];
    for (int j = jb; j < je; ++j) {
        int   s   = csr_src[j];
        float nrm = csr_nrm[j];
        const float4 v = ((const float4*)(xw + (size_t)s * FDIM))[lane];
        acc.x = fmaf(v.x, nrm, acc.x);
        acc.y = fmaf(v.y, nrm, acc.y);
        acc.z = fmaf(v.z, nrm, acc.z);
        acc.w = fmaf(v.w, nrm, acc.w);
    }

    const float4 bv = ((const float4*)bias)[lane];
    acc.x += bv.x; acc.y += bv.y; acc.z += bv.z; acc.w += bv.w;
    if (do_elu) {
        acc.x = (acc.x > 0.0f) ? acc.x : expm1f(acc.x);
        acc.y = (acc.y > 0.0f) ? acc.y : expm1f(acc.y);
        acc.z = (acc.z > 0.0f) ? acc.z : expm1f(acc.z);
        acc.w = (acc.w > 0.0f) ? acc.w : expm1f(acc.w);
    }
    ((float4*)(h + (size_t)node * FDIM))[lane] = acc;
}

// ---------------------------------------------------------------------------
// Pooling over sorted batch: block = 128 feature-threads covering 32 nodes;
// flush atomic only at graph boundaries (~32x atomic reduction).
// ---------------------------------------------------------------------------
__global__ void __launch_bounds__(128)
pool_kernel(const float* __restrict__ h, const int* __restrict__ batch,
            float* __restrict__ pooled, int n) {
    int f = threadIdx.x;
    int start = blockIdx.x * 32;
    int end = min(start + 32, n);
    if (start >= n) return;
    float acc = 0.0f;
    int cur = batch[start];
    for (int i = start; i < end; ++i) {
        int bg = batch[i];
        if (bg != cur) {
            atomAddF(&pooled[(size_t)cur * FDIM + f], acc);
            acc = 0.0f;
            cur = bg;
        }
        acc += h[(size_t)i * FDIM + f];
    }
    atomAddF(&pooled[(size_t)cur * FDIM + f], acc);
}

__global__ void graph_count_kernel(const int* __restrict__ batch, float* __restrict__ cnt, int n) {
    int i = blockIdx.x * blockDim.x + threadIdx.x;
    if (i < n) atomAddF(&cnt[batch[i]], 1.0f);
}

// ---------------------------------------------------------------------------
// Head MLP: one 32-thread block per graph.
// ---------------------------------------------------------------------------
__global__ void __launch_bounds__(32)
mlp_kernel(const float* __restrict__ pooled, const float* __restrict__ cnt,
           const float* __restrict__ stats,
           const float* __restrict__ fw1, const float* __restrict__ fb1,
           const float* __restrict__ fw2, const float* __restrict__ fb2,
           const float* __restrict__ fw3, const float* __restrict__ fb3,
           float* __restrict__ out) {
    const int g = blockIdx.x;
    const int j = threadIdx.x;
    __shared__ float gv[FDIM + GF];
    __shared__ float h1[32];
    __shared__ float h2[16];

    float c = fmaxf(cnt[g], 1.0f);
    for (int k = j; k < FDIM; k += 32) gv[k] = pooled[(size_t)g * FDIM + k] / c;
    if (j < GF) gv[FDIM + j] = stats[g * GF + j];
    __syncthreads();

    float a = fb1[j];
    for (int k = 0; k < FDIM + GF; ++k) a += gv[k] * fw1[k * 32 + j];
    h1[j] = fmaxf(a, 0.0f);
    __syncthreads();

    if (j < 16) {
        float a2 = fb2[j];
        for (int k = 0; k < 32; ++k) a2 += h1[k] * fw2[k * 16 + j];
        h2[j] = fmaxf(a2, 0.0f);
    }
    __syncthreads();

    if (j == 0) {
        float a3 = fb3[0];
        for (int k = 0; k < 16; ++k) a3 += h2[k] * fw3[k];
        out[g] = a3;
    }
}

// ---------------------------------------------------------------------------
// Launch
// ---------------------------------------------------------------------------
extern "C" void kernel_launch(void* const* d_in, const int* in_sizes, int n_in,
                              void* d_out, int out_size, void* d_ws, size_t ws_size,
                              hipStream_t stream) {
    const float* x     = (const float*)d_in[0];
    const int*   ei    = (const int*)d_in[1];
    const int*   batch = (const int*)d_in[2];
    const float* stats = (const float*)d_in[3];
    const float* W1 = (const float*)d_in[4];
    const float* b1 = (const float*)d_in[5];
    const float* W2 = (const float*)d_in[6];
    const float* b2 = (const float*)d_in[7];
    const float* W3 = (const float*)d_in[8];
    const float* b3 = (const float*)d_in[9];
    const float* fw1 = (const float*)d_in[10];
    const float* fb1 = (const float*)d_in[11];
    const float* fw2 = (const float*)d_in[12];
    const float* fb2 = (const float*)d_in[13];
    const float* fw3 = (const float*)d_in[14];
    const float* fb3 = (const float*)d_in[15];

    const int* src = ei;
    const int* dst = ei + N_EDGES;

    // --- workspace carve-out (256B aligned chunks) ---
    char* ws = (char*)d_ws;
    size_t off = 0;
    auto carve = [&](size_t bytes) -> void* {
        void* p = ws + off;
        off += (bytes + 255) & ~(size_t)255;
        return p;
    };
    float* xw      = (float*)carve((size_t)N_NODES * FDIM * 4);
    float* h       = (float*)carve((size_t)N_NODES * FDIM * 4);
    float* dinv    = (float*)carve((size_t)N_NODES * 4);
    int*   counts  = (int*)  carve((size_t)N_NODES * 4);
    int*   offsets = (int*)  carve((size_t)(N_NODES + 1) * 4);
    int*   cursor  = (int*)  carve((size_t)N_NODES * 4);
    int*   csr_src = (int*)  carve((size_t)N_EDGES * 4);
    float* csr_nrm = (float*)carve((size_t)N_EDGES * 4);
    float* pooled  = (float*)carve((size_t)N_GRAPHS * FDIM * 4);
    float* cnt     = (float*)carve((size_t)N_GRAPHS * 4);

    const int NTOT = N_NODES * FDIM;
    (void)NTOT;

    // --- build dinv + destination-CSR (once; amortized over 3 layers) ---
    zeroi_kernel<<<(N_NODES + 255) / 256, 256, 0, stream>>>(counts, N_NODES);
    deg_count_kernel<<<(N_EDGES + 255) / 256, 256, 0, stream>>>(dst, counts, N_EDGES);
    dinv_kernel<<<(N_NODES + 255) / 256, 256, 0, stream>>>(counts, dinv, N_NODES);
    scan_kernel<<<1, 1024, 0, stream>>>(counts, offsets, cursor, N_NODES);
    csr_build_kernel<<<(N_EDGES + 255) / 256, 256, 0, stream>>>(src, dst, dinv, cursor,
                                                                csr_src, csr_nrm, N_EDGES);

    // --- 3 GCN layers: WMMA GEMM + fused atomic-free aggregation ---
    const float* Ws[3] = {W1, W2, W3};
    const float* bs[3] = {b1, b2, b3};
    const float* cur = x;
    const int agg_blocks = (N_NODES * 32 + 255) / 256;   // 12500

    for (int l = 0; l < 3; ++l) {
        gemm128_wmma_kernel<<<N_NODES / 16, 256, 0, stream>>>(cur, Ws[l], xw);
        agg_fused_kernel<<<agg_blocks, 256, 0, stream>>>(xw, offsets, csr_src, csr_nrm,
                                                         dinv, bs[l], h, l < 2 ? 1 : 0,
                                                         N_NODES);
        cur = h;
    }

    // --- pooling ---
    zerof_kernel<<<(N_GRAPHS * FDIM + N_GRAPHS + 255) / 256, 256, 0, stream>>>(
        pooled, N_GRAPHS * FDIM + N_GRAPHS);
    graph_count_kernel<<<(N_NODES + 255) / 256, 256, 0, stream>>>(batch, cnt, N_NODES);
    pool_kernel<<<(N_NODES + 31) / 32, 128, 0, stream>>>(h, batch, pooled, N_NODES);

    // --- head MLP ---
    mlp_kernel<<<N_GRAPHS, 32, 0, stream>>>(pooled, cnt, stats,
                                            fw1, fb1, fw2, fb2, fw3, fb3,
                                            (float*)d_out);
}